// HyperGCN_54941221650615
// MI455X (gfx1250) — compile-verified
//
#include <hip/hip_runtime.h>
#include <hip/hip_bf16.h>

// ---------------- problem constants ----------------
#define BB 32
#define LL 25
#define D_IN 1024
#define HH 2048
#define NNODE (BB * LL)        // 800
#define NN3 (3 * NNODE)        // 2400
#define D2 (D_IN + 2 * HH)     // 5120
#define NUM_L 3
#define NUM_K 4

typedef float v2f __attribute__((ext_vector_type(2)));
typedef float v8f __attribute__((ext_vector_type(8)));

// ---------------- elementwise / setup kernels ----------------

__global__ void zero_kernel(float* __restrict__ p, size_t n) {
  size_t i = (size_t)blockIdx.x * blockDim.x + threadIdx.x;
  size_t stride = (size_t)gridDim.x * blockDim.x;
  for (; i < n; i += stride) p[i] = 0.0f;
}

// feats[row, d], row = b*3L + m*L + t;  m: 0->l(+spk emb), 1->a, 2->v
__global__ void build_feats_kernel(const float* __restrict__ a,
                                   const float* __restrict__ v,
                                   const float* __restrict__ l,
                                   const float* __restrict__ qmask,   // (L,B,2)
                                   const float* __restrict__ spk_emb, // (2,D_IN)
                                   float* __restrict__ feats) {
  size_t idx = (size_t)blockIdx.x * blockDim.x + threadIdx.x;
  size_t total = (size_t)NN3 * D_IN;
  if (idx >= total) return;
  int row = (int)(idx / D_IN);
  int d = (int)(idx % D_IN);
  int b = row / (3 * LL);
  int m = (row / LL) % 3;
  int t = row % LL;
  int src = b * LL + t;
  float val;
  if (m == 0) {
    float q0 = qmask[((size_t)t * BB + b) * 2 + 0];
    float q1 = qmask[((size_t)t * BB + b) * 2 + 1];
    int spk = (q1 > q0) ? 1 : 0;
    val = l[(size_t)src * D_IN + d] + spk_emb[(size_t)spk * D_IN + d];
  } else if (m == 1) {
    val = a[(size_t)src * D_IN + d];
  } else {
    val = v[(size_t)src * D_IN + d];
  }
  feats[idx] = val;
}

__global__ void edge_attr_kernel(const float* __restrict__ attr1,
                                 const float* __restrict__ attr2,
                                 const float* __restrict__ htype,
                                 float* __restrict__ ea, int E) {
  size_t idx = (size_t)blockIdx.x * blockDim.x + threadIdx.x;
  if (idx >= (size_t)E * HH) return;
  int e = (int)(idx / HH);
  int h = (int)(idx % HH);
  float ht = htype[e];
  ea[idx] = attr1[h] * ht + attr2[h] * (1.0f - ht);
}

// Be[edge] += 1 ; De[node] += w[edge]
__global__ void counts_kernel(const int* __restrict__ nidx,
                              const int* __restrict__ eidx,
                              const float* __restrict__ w,
                              float* __restrict__ Be, float* __restrict__ De,
                              int NHE) {
  int j = blockIdx.x * blockDim.x + threadIdx.x;
  if (j >= NHE) return;
  int e = eidx[j];
  int n = nidx[j];
  atomicAdd(&Be[e], 1.0f);
  atomicAdd(&De[n], w[e]);
}

__global__ void gnn_deg_kernel(const int* __restrict__ dst, float* __restrict__ deg, int NGE) {
  int e = blockIdx.x * blockDim.x + threadIdx.x;
  if (e >= NGE) return;
  atomicAdd(&deg[dst[e]], 1.0f);
}

__global__ void gnn_norm_kernel(const int* __restrict__ src, const int* __restrict__ dst,
                                const float* __restrict__ deg, float* __restrict__ norm, int NGE) {
  int e = blockIdx.x * blockDim.x + threadIdx.x;
  if (e >= NGE) return;
  float ds = deg[src[e]];
  float dd = deg[dst[e]];
  float is = (ds > 0.0f) ? rsqrtf(ds) : 0.0f;
  float id = (dd > 0.0f) ? rsqrtf(dd) : 0.0f;
  norm[e] = is * id;
}

// ---------------- WMMA f32 GEMM: C[M,Nc] = A[M,K] @ W[K,Nc] (+bias) ----------------
// 64x64 tile per 256-thread block (8 wave32), LDS-staged K in chunks of 32.
#define TM 64
#define TN 64
#define TK 32

__global__ __launch_bounds__(256)
void gemm_wmma_f32(const float* __restrict__ A, const float* __restrict__ W,
                   const float* __restrict__ bias, float* __restrict__ C,
                   int M, int K, int Nc) {
  __shared__ float lA[TM][TK + 1];  // +1 pad: kills 32-float bank-conflict stride
  __shared__ float lB[TK][TN];

  const int tile_n = blockIdx.x * TN;
  const int tile_m = blockIdx.y * TM;
  const int tid = threadIdx.x;
  const int wave = tid >> 5;
  const int lane = tid & 31;
  const int half = lane >> 4;   // 0: lanes 0-15, 1: lanes 16-31
  const int l16 = lane & 15;

  const int mi = wave >> 1;            // 0..3 : which 16-row band
  const int nj0 = (wave & 1) * 2;      // 0 or 2: first of two 16-col bands

  v8f acc0 = {};
  v8f acc1 = {};

  for (int k0 = 0; k0 < K; k0 += TK) {
    // stage A 64x32 (zero-pad M edge so WMMA runs with EXEC all-ones)
    for (int i = tid; i < TM * TK; i += 256) {
      int r = i >> 5;
      int c = i & 31;
      int gm = tile_m + r;
      lA[r][c] = (gm < M) ? A[(size_t)gm * K + (k0 + c)] : 0.0f;
    }
    // stage B 32x64
    for (int i = tid; i < TK * TN; i += 256) {
      int r = i >> 6;
      int c = i & 63;
      lB[r][c] = W[(size_t)(k0 + r) * Nc + (tile_n + c)];
    }
    __syncthreads();

#pragma unroll
    for (int kk = 0; kk < TK; kk += 4) {
      // A fragment (16x4 f32): v0 = K=0 (lanes 0-15) / K=2 (lanes 16-31); v1 = K+1
      int ka = kk + 2 * half;
      v2f afrag;
      afrag.x = lA[mi * 16 + l16][ka];
      afrag.y = lA[mi * 16 + l16][ka + 1];
      // B fragments (4x16 f32), same half-split over K
      v2f b0, b1;
      b0.x = lB[ka][nj0 * 16 + l16];
      b0.y = lB[ka + 1][nj0 * 16 + l16];
      b1.x = lB[ka][(nj0 + 1) * 16 + l16];
      b1.y = lB[ka + 1][(nj0 + 1) * 16 + l16];
      acc0 = __builtin_amdgcn_wmma_f32_16x16x4_f32(false, afrag, false, b0,
                                                   (short)0, acc0, false, false);
      acc1 = __builtin_amdgcn_wmma_f32_16x16x4_f32(false, afrag, false, b1,
                                                   (short)0, acc1, false, false);
    }
    __syncthreads();
  }

  // D layout: VGPR r -> lanes0-15: M=r, lanes16-31: M=r+8; N = l16 within band
#pragma unroll
  for (int r = 0; r < 8; ++r) {
    int gm = tile_m + mi * 16 + r + half * 8;
    if (gm < M) {
      int gn0 = tile_n + nj0 * 16 + l16;
      int gn1 = gn0 + 16;
      float v0 = acc0[r] + (bias ? bias[gn0] : 0.0f);
      float v1 = acc1[r] + (bias ? bias[gn1] : 0.0f);
      C[(size_t)gm * Nc + gn0] = v0;
      C[(size_t)gm * Nc + gn1] = v1;
    }
  }
}

// ---------------- hyper-conv scatter kernels ----------------

// he[eidx[j]] += xl[nidx[j]] * ew[j]      grid: (NHE, H/256)
__global__ void scatter_he_kernel(const float* __restrict__ xl,
                                  const int* __restrict__ nidx,
                                  const int* __restrict__ eidx,
                                  const float* __restrict__ ew,
                                  float* __restrict__ he) {
  int j = blockIdx.x;
  int h = blockIdx.y * blockDim.x + threadIdx.x;
  atomicAdd(&he[(size_t)eidx[j] * HH + h], xl[(size_t)nidx[j] * HH + h] * ew[j]);
}

// he = he / max(Be,1-ified) + el
__global__ void he_finalize_kernel(float* __restrict__ he, const float* __restrict__ el,
                                   const float* __restrict__ Be, int E) {
  size_t idx = (size_t)blockIdx.x * blockDim.x + threadIdx.x;
  if (idx >= (size_t)E * HH) return;
  int e = (int)(idx / HH);
  float cnt = Be[e];
  cnt = (cnt > 0.0f) ? cnt : 1.0f;
  he[idx] = he[idx] / cnt + el[idx];
}

// nxt[nidx[j]] += he[eidx[j]] * w[eidx[j]]   grid: (NHE, H/256)
__global__ void scatter_out_kernel(const float* __restrict__ he,
                                   const int* __restrict__ nidx,
                                   const int* __restrict__ eidx,
                                   const float* __restrict__ w,
                                   float* __restrict__ nxt) {
  int j = blockIdx.x;
  int h = blockIdx.y * blockDim.x + threadIdx.x;
  int e = eidx[j];
  atomicAdd(&nxt[(size_t)nidx[j] * HH + h], he[(size_t)e * HH + h] * w[e]);
}

// nxt = nxt / De' + b
__global__ void out_finalize_kernel(float* __restrict__ nxt, const float* __restrict__ De,
                                    const float* __restrict__ b) {
  size_t idx = (size_t)blockIdx.x * blockDim.x + threadIdx.x;
  if (idx >= (size_t)NN3 * HH) return;
  int n = (int)(idx / HH);
  int h = (int)(idx % HH);
  float d = De[n];
  d = (d > 0.0f) ? d : 1.0f;
  nxt[idx] = nxt[idx] / d + b[h];
}

// g[dst[e]] += h[src[e]] * norm[e]    grid: (NGE, H/256)
__global__ void gnn_scatter_kernel(const float* __restrict__ hbuf,
                                   const int* __restrict__ src,
                                   const int* __restrict__ dst,
                                   const float* __restrict__ norm,
                                   float* __restrict__ g) {
  int e = blockIdx.x;
  int h = blockIdx.y * blockDim.x + threadIdx.x;
  atomicAdd(&g[(size_t)dst[e] * HH + h], hbuf[(size_t)src[e] * HH + h] * norm[e]);
}

// out[n, m*D2 + d] = {feats | hyper_out | g}[b*3L + m*L + t, d]
__global__ void final_concat_kernel(const float* __restrict__ feats,
                                    const float* __restrict__ hout,
                                    const float* __restrict__ g,
                                    float* __restrict__ out) {
  size_t idx = (size_t)blockIdx.x * blockDim.x + threadIdx.x;
  size_t total = (size_t)NNODE * 3 * D2;
  if (idx >= total) return;
  int d = (int)(idx % D2);
  int m = (int)((idx / D2) % 3);
  int n = (int)(idx / ((size_t)3 * D2));
  int b = n / LL;
  int t = n % LL;
  int row = b * 3 * LL + m * LL + t;
  float val;
  if (d < D_IN) val = feats[(size_t)row * D_IN + d];
  else if (d < D_IN + HH) val = hout[(size_t)row * HH + (d - D_IN)];
  else val = g[(size_t)row * HH + (d - D_IN - HH)];
  out[(size_t)n * (3 * D2) + (size_t)m * D2 + d] = val;
}

// ---------------- host launcher ----------------

static inline void launch_zero(float* p, size_t n, hipStream_t s) {
  int blocks = (int)((n + 255) / 256);
  if (blocks > 65535) blocks = 65535;
  zero_kernel<<<blocks, 256, 0, s>>>(p, n);
}

extern "C" void kernel_launch(void* const* d_in, const int* in_sizes, int n_in,
                              void* d_out, int out_size, void* d_ws, size_t ws_size,
                              hipStream_t stream) {
  const float* a_in   = (const float*)d_in[0];
  const float* v_in   = (const float*)d_in[1];
  const float* l_in   = (const float*)d_in[2];
  const float* qmask  = (const float*)d_in[3];
  const float* spkemb = (const float*)d_in[4];
  const float* W_fc1  = (const float*)d_in[5];
  const float* b_fc1  = (const float*)d_in[6];
  const float* hew    = (const float*)d_in[7];
  const float* EW     = (const float*)d_in[8];
  const float* attr1  = (const float*)d_in[9];
  const float* attr2  = (const float*)d_in[10];
  const float* W_hyp  = (const float*)d_in[11];
  const float* b_hyp  = (const float*)d_in[12];
  const float* W_gnn  = (const float*)d_in[13];
  const float* b_gnn  = (const float*)d_in[14];
  const int*   nidx   = (const int*)d_in[15];
  const int*   eidx   = (const int*)d_in[16];
  const float* htype  = (const float*)d_in[17];
  const int*   gedge  = (const int*)d_in[18];

  const int NHE = in_sizes[15];        // 4800
  const int E   = in_sizes[17];        // 896
  const int NGE = in_sizes[18] / 2;    // 62400
  const int* gsrc = gedge;
  const int* gdst = gedge + NGE;

  // ---- workspace layout (floats) ----
  float* ws = (float*)d_ws;
  size_t off = 0;
  float* feats = ws + off; off += (size_t)NN3 * D_IN;   // 2,457,600
  float* x1    = ws + off; off += (size_t)NN3 * HH;
  float* bufA  = ws + off; off += (size_t)NN3 * HH;
  float* bufB  = ws + off; off += (size_t)NN3 * HH;
  float* xl    = ws + off; off += (size_t)NN3 * HH;
  float* g     = ws + off; off += (size_t)NN3 * HH;
  float* hbuf  = ws + off; off += (size_t)NN3 * HH;
  float* eattr = ws + off; off += (size_t)E * HH;
  float* el    = ws + off; off += (size_t)E * HH;
  float* he    = ws + off; off += (size_t)E * HH;
  float* Be    = ws + off; off += (size_t)E;
  float* De    = ws + off; off += (size_t)NN3;
  float* deg   = ws + off; off += (size_t)NN3;
  float* norm  = ws + off; off += (size_t)NGE;
  (void)ws_size; (void)n_in; (void)out_size;

  // ---- setup ----
  {
    size_t tot = (size_t)NN3 * D_IN;
    build_feats_kernel<<<(unsigned)((tot + 255) / 256), 256, 0, stream>>>(
        a_in, v_in, l_in, qmask, spkemb, feats);
  }
  {
    size_t tot = (size_t)E * HH;
    edge_attr_kernel<<<(unsigned)((tot + 255) / 256), 256, 0, stream>>>(
        attr1, attr2, htype, eattr, E);
  }
  launch_zero(Be, E, stream);
  launch_zero(De, NN3, stream);
  launch_zero(deg, NN3, stream);
  counts_kernel<<<(NHE + 255) / 256, 256, 0, stream>>>(nidx, eidx, hew, Be, De, NHE);
  gnn_deg_kernel<<<(NGE + 255) / 256, 256, 0, stream>>>(gdst, deg, NGE);
  gnn_norm_kernel<<<(NGE + 255) / 256, 256, 0, stream>>>(gsrc, gdst, deg, norm, NGE);

  // ---- x1 = feats @ W_fc1 + b_fc1 ----
  {
    dim3 grid(HH / TN, (NN3 + TM - 1) / TM);
    gemm_wmma_f32<<<grid, 256, 0, stream>>>(feats, W_fc1, b_fc1, x1, NN3, D_IN, HH);
  }

  // seed GNN chain: g = x1
  hipMemcpyAsync(g, x1, (size_t)NN3 * HH * sizeof(float),
                 hipMemcpyDeviceToDevice, stream);

  // ---- 3 hyper-conv layers ----
  const float* cur = x1;
  float* nxt = bufA;
  for (int i = 0; i < NUM_L; ++i) {
    const float* Wl = W_hyp + (size_t)i * HH * HH;
    const float* bl = b_hyp + (size_t)i * HH;
    {
      dim3 grid(HH / TN, (NN3 + TM - 1) / TM);
      gemm_wmma_f32<<<grid, 256, 0, stream>>>(cur, Wl, nullptr, xl, NN3, HH, HH);
    }
    {
      dim3 grid(HH / TN, (E + TM - 1) / TM);
      gemm_wmma_f32<<<grid, 256, 0, stream>>>(eattr, Wl, nullptr, el, E, HH, HH);
    }
    launch_zero(he, (size_t)E * HH, stream);
    {
      dim3 grid(NHE, HH / 256);
      scatter_he_kernel<<<grid, 256, 0, stream>>>(xl, nidx, eidx, EW, he);
    }
    {
      size_t tot = (size_t)E * HH;
      he_finalize_kernel<<<(unsigned)((tot + 255) / 256), 256, 0, stream>>>(he, el, Be, E);
    }
    launch_zero(nxt, (size_t)NN3 * HH, stream);
    {
      dim3 grid(NHE, HH / 256);
      scatter_out_kernel<<<grid, 256, 0, stream>>>(he, nidx, eidx, hew, nxt);
    }
    {
      size_t tot = (size_t)NN3 * HH;
      out_finalize_kernel<<<(unsigned)((tot + 255) / 256), 256, 0, stream>>>(nxt, De, bl);
    }
    cur = nxt;
    nxt = (i == 0) ? bufB : bufA;
  }
  const float* hyper_out = cur;  // bufA after 3 layers (x1 -> A -> B -> A)

  // ---- 4 GNN layers: h = g@W + b ; g += scatter(h[src]*norm -> dst) ----
  for (int k = 0; k < NUM_K; ++k) {
    const float* Wk = W_gnn + (size_t)k * HH * HH;
    const float* bk = b_gnn + (size_t)k * HH;
    {
      dim3 grid(HH / TN, (NN3 + TM - 1) / TM);
      gemm_wmma_f32<<<grid, 256, 0, stream>>>(g, Wk, bk, hbuf, NN3, HH, HH);
    }
    {
      dim3 grid(NGE, HH / 256);
      gnn_scatter_kernel<<<grid, 256, 0, stream>>>(hbuf, gsrc, gdst, norm, g);
    }
  }

  // ---- final assembly ----
  {
    size_t tot = (size_t)NNODE * 3 * D2;
    final_concat_kernel<<<(unsigned)((tot + 255) / 256), 256, 0, stream>>>(
        feats, hyper_out, g, (float*)d_out);
  }
}